// LlamaAttention_68702296867555
// MI455X (gfx1250) — compile-verified
//
#include <hip/hip_runtime.h>
#include <hip/hip_bf16.h>
#include <math.h>

// ---------------- problem constants (match reference) ----------------
constexpr int BATCH   = 16;
constexpr int HIDDEN  = 4096;
constexpr int NQ      = 32;
constexpr int NKV     = 8;
constexpr int HD      = 128;
constexpr int CTX_MAX = 4096;
constexpr int KVBLOCK = 16;
constexpr int NBLK    = CTX_MAX / KVBLOCK;       // 256
constexpr int QKV_N   = (NQ + 2 * NKV) * HD;     // 6144
constexpr int GRP     = NQ / NKV;                // 4 query heads per kv head

typedef __attribute__((ext_vector_type(2))) float v2f;
typedef __attribute__((ext_vector_type(8))) float v8f;

// =====================================================================
// Kernel 1/4: skinny GEMM  Y[16,N] = X[16,K] * W[K,N]   (fp32 WMMA)
// One 16x16 output tile per wave; 4 waves (128 threads) per block.
// =====================================================================
__global__ __launch_bounds__(128)
void gemm16_wmma(const float* __restrict__ X, const float* __restrict__ W,
                 float* __restrict__ Y, int N, int K) {
    const int lane = threadIdx.x & 31;
    const int wave = threadIdx.x >> 5;
    const int tile = blockIdx.x * 4 + wave;
    const int n0   = tile * 16;
    if (n0 >= N) return;                      // wave-uniform

    const int m    = lane & 15;               // A row / D column index
    const int koff = (lane >> 4) * 2;         // upper half-wave covers K+2,K+3

    const float* __restrict__ xrow = X + (size_t)m * K;
    v8f acc = {};

    for (int k = 0; k < K; k += 4) {
        v2f a, b;
        a.x = xrow[k + koff];
        a.y = xrow[k + koff + 1];
        const float* wp = W + (size_t)(k + koff) * N + n0 + m;
        b.x = wp[0];
        b.y = wp[N];
        acc = __builtin_amdgcn_wmma_f32_16x16x4_f32(
                  false, a, false, b, (short)0, acc, false, false);
    }

    // D layout: VGPR r -> row r (lanes 0-15) / row r+8 (lanes 16-31), col = lane&15
    const int rbase = (lane < 16) ? 0 : 8;
    const int col   = n0 + m;
#pragma unroll
    for (int r = 0; r < 8; ++r)
        Y[(size_t)(rbase + r) * N + col] = acc[r];
}

// =====================================================================
// Kernel 2: RoPE on current-token q,k ; copy v.  One block per batch.
// qkv row layout: [q: NQ*HD | k: NKV*HD | v: NKV*HD]
// =====================================================================
__global__ __launch_bounds__(128)
void rope_kernel(const float* __restrict__ qkv, const int* __restrict__ ctx,
                 float* __restrict__ qr, float* __restrict__ kr,
                 float* __restrict__ vc) {
    const int b = blockIdx.x;
    int pos = ctx[b] - 1;
    pos = pos < 0 ? 0 : (pos > CTX_MAX - 1 ? CTX_MAX - 1 : pos);
    const float fpos = (float)pos;
    const float* row = qkv + (size_t)b * QKV_N;

    // rotate q and k: (NQ+NKV)*HD/2 pairs
    for (int idx = threadIdx.x; idx < (NQ + NKV) * (HD / 2); idx += blockDim.x) {
        const int head = idx >> 6;            // /64
        const int i    = idx & 63;
        const float inv = __powf(10000.0f, -(float)i / 64.0f);
        float s, c;
        __sincosf(fpos * inv, &s, &c);
        const float* src;
        float* dst;
        if (head < NQ) { src = row + head * HD;            dst = qr + ((size_t)b * NQ + head) * HD; }
        else { const int kh = head - NQ;
               src = row + NQ * HD + kh * HD;              dst = kr + ((size_t)b * NKV + kh) * HD; }
        const float x1 = src[i], x2 = src[i + 64];
        dst[i]      = x1 * c - x2 * s;
        dst[i + 64] = x2 * c + x1 * s;
    }
    // copy v
    for (int idx = threadIdx.x; idx < NKV * HD; idx += blockDim.x)
        vc[(size_t)b * NKV * HD + idx] = row[(NQ + NKV) * HD + idx];
}

// =====================================================================
// Kernel 3: flash-decode paged attention.
// grid = BATCH*NKV blocks, 128 threads (4 waves). Waves round-robin over
// 16-token KV pages; per page: stage RoPE'd K in LDS, WMMA score tile
// (16 padded heads x 16 tokens), online softmax, P*V accumulate.
// =====================================================================
__global__ __launch_bounds__(128)
void attn_kernel(const float* __restrict__ qr, const float* __restrict__ kr,
                 const float* __restrict__ vcur,
                 const float* __restrict__ key_cache,
                 const float* __restrict__ value_cache,
                 const int* __restrict__ block_tables,
                 const int* __restrict__ ctx,
                 float* __restrict__ attn_out) {
    constexpr int NW = 4;                      // waves per block
    __shared__ float q_s[16][HD];              // padded query group (rows 4..15 zero)
    __shared__ float k_s[NW][KVBLOCK][HD];     // per-wave rotated K page
    __shared__ float p_s[NW][GRP][KVBLOCK];    // per-wave probabilities
    __shared__ float red[NW][GRP][2];          // per-wave (m, l)
    __shared__ float acc_s[NW][GRP][HD];       // per-wave partial outputs

    const int b    = blockIdx.x / NKV;
    const int kvh  = blockIdx.x % NKV;
    const int lane = threadIdx.x & 31;
    const int wave = threadIdx.x >> 5;

    const int ctxlen = ctx[b];
    int pos = ctxlen - 1;
    pos = pos < 0 ? 0 : (pos > CTX_MAX - 1 ? CTX_MAX - 1 : pos);

    // load (padded) rotated query group
    for (int idx = threadIdx.x; idx < 16 * HD; idx += blockDim.x) {
        const int h = idx >> 7, d = idx & 127;
        q_s[h][d] = (h < GRP) ? qr[((size_t)b * NQ + kvh * GRP + h) * HD + d] : 0.0f;
    }
    __syncthreads();

    const float scale = 0.08838834764831845f;  // 128^-0.5
    float m_h[GRP], l_h[GRP], acc[GRP][4];
#pragma unroll
    for (int h = 0; h < GRP; ++h) {
        m_h[h] = -INFINITY; l_h[h] = 0.0f;
#pragma unroll
        for (int j = 0; j < 4; ++j) acc[h][j] = 0.0f;
    }

    const int npages = (ctxlen + KVBLOCK - 1) / KVBLOCK;
    const int mm   = lane & 15;
    const int koff = (lane >> 4) * 2;

    for (int page = wave; page < npages; page += NW) {
        const int s0  = page * KVBLOCK;
        const int blk = block_tables[b * NBLK + page];
        const float* __restrict__ kc =
            key_cache + (((size_t)blk * NKV + kvh) * KVBLOCK) * HD;

        // ---- stage RoPE'd K page into LDS (current token replaces slot pos)
#pragma unroll
        for (int j = 0; j < 2; ++j) {
            const int d = lane + 32 * j;       // pair index in [0,64)
            const float inv = __powf(10000.0f, -(float)d / 64.0f);
            for (int t = 0; t < KVBLOCK; ++t) {
                const int s = s0 + t;
                if (s == pos) {                // wave-uniform branch
                    k_s[wave][t][d]      = kr[((size_t)b * NKV + kvh) * HD + d];
                    k_s[wave][t][d + 64] = kr[((size_t)b * NKV + kvh) * HD + d + 64];
                } else {
                    const float x1 = kc[t * HD + d];
                    const float x2 = kc[t * HD + d + 64];
                    float sn, c;
                    __sincosf((float)s * inv, &sn, &c);
                    k_s[wave][t][d]      = x1 * c - x2 * sn;
                    k_s[wave][t][d + 64] = x2 * c + x1 * sn;
                }
            }
        }

        // ---- score tile: D[head(pad16) x token16] = Q * K^T, via fp32 WMMA
        v8f sc = {};
        for (int kk = 0; kk < HD; kk += 4) {
            v2f a, bb;
            a.x  = q_s[mm][kk + koff];
            a.y  = q_s[mm][kk + koff + 1];
            bb.x = k_s[wave][mm][kk + koff];      // B[K=d][N=token]
            bb.y = k_s[wave][mm][kk + koff + 1];
            sc = __builtin_amdgcn_wmma_f32_16x16x4_f32(
                     false, a, false, bb, (short)0, sc, false, false);
        }

        // ---- online softmax per head (head h lives in sc[h], lanes 0-15)
#pragma unroll
        for (int h = 0; h < GRP; ++h) {
            float v = sc[h] * scale;
            const int t = lane & 15;
            if (lane >= 16 || (s0 + t) >= ctxlen) v = -1e30f;
            float mx = v;
#pragma unroll
            for (int off = 16; off >= 1; off >>= 1)
                mx = fmaxf(mx, __shfl_xor(mx, off, 32));
            const float m_new = fmaxf(m_h[h], mx);
            const float p     = __expf(v - m_new);
            float psum = p;
#pragma unroll
            for (int off = 16; off >= 1; off >>= 1)
                psum += __shfl_xor(psum, off, 32);
            const float corr = __expf(m_h[h] - m_new);
            l_h[h] = l_h[h] * corr + psum;
            m_h[h] = m_new;
#pragma unroll
            for (int j = 0; j < 4; ++j) acc[h][j] *= corr;
            if (lane < 16) p_s[wave][h][lane] = p;
        }

        // ---- P * V accumulate (d = lane + 32j, coalesced 128B per j)
        const float* __restrict__ vcache =
            value_cache + (((size_t)blk * NKV + kvh) * KVBLOCK) * HD;
        const int tmax = (ctxlen - s0 < KVBLOCK) ? (ctxlen - s0) : KVBLOCK;
        for (int t = 0; t < tmax; ++t) {
            const int s = s0 + t;
            float pv[GRP];
#pragma unroll
            for (int h = 0; h < GRP; ++h) pv[h] = p_s[wave][h][t];
#pragma unroll
            for (int j = 0; j < 4; ++j) {
                const int d = lane + 32 * j;
                const float vv = (s == pos)
                    ? vcur[((size_t)b * NKV + kvh) * HD + d]
                    : vcache[t * HD + d];
#pragma unroll
                for (int h = 0; h < GRP; ++h)
                    acc[h][j] = fmaf(pv[h], vv, acc[h][j]);
            }
        }
    }

    // ---- cross-wave merge
    if (lane == 0) {
#pragma unroll
        for (int h = 0; h < GRP; ++h) { red[wave][h][0] = m_h[h]; red[wave][h][1] = l_h[h]; }
    }
#pragma unroll
    for (int h = 0; h < GRP; ++h)
#pragma unroll
        for (int j = 0; j < 4; ++j)
            acc_s[wave][h][lane + 32 * j] = acc[h][j];
    __syncthreads();

    const int h = wave;                         // each wave finalizes one head
    float gm = -INFINITY;
#pragma unroll
    for (int w = 0; w < NW; ++w) gm = fmaxf(gm, red[w][h][0]);
    float gl = 0.0f;
#pragma unroll
    for (int w = 0; w < NW; ++w) gl += red[w][h][1] * __expf(red[w][h][0] - gm);
    const float inv_gl = 1.0f / gl;
#pragma unroll
    for (int j = 0; j < 4; ++j) {
        const int d = lane + 32 * j;
        float s = 0.0f;
#pragma unroll
        for (int w = 0; w < NW; ++w)
            s += acc_s[w][h][d] * __expf(red[w][h][0] - gm);
        attn_out[((size_t)b * NQ + kvh * GRP + h) * HD + d] = s * inv_gl;
    }
}

// =====================================================================
// launcher
// =====================================================================
extern "C" void kernel_launch(void* const* d_in, const int* in_sizes, int n_in,
                              void* d_out, int out_size, void* d_ws, size_t ws_size,
                              hipStream_t stream) {
    const float* hidden       = (const float*)d_in[0];
    const float* key_cache    = (const float*)d_in[1];
    const float* value_cache  = (const float*)d_in[2];
    const float* W_qkv        = (const float*)d_in[3];
    const float* W_o          = (const float*)d_in[4];
    const int*   block_tables = (const int*)d_in[5];
    const int*   context_lens = (const int*)d_in[6];
    float* out = (float*)d_out;

    float* qkv  = (float*)d_ws;                          // 16*6144
    float* qr   = qkv  + (size_t)BATCH * QKV_N;          // 16*32*128
    float* kr   = qr   + (size_t)BATCH * NQ * HD;        // 16*8*128
    float* vcur = kr   + (size_t)BATCH * NKV * HD;       // 16*8*128
    float* attn = vcur + (size_t)BATCH * NKV * HD;       // 16*4096

    // 1) QKV projection: N=6144 -> 384 tiles -> 96 blocks of 4 waves
    gemm16_wmma<<<QKV_N / 64, 128, 0, stream>>>(hidden, W_qkv, qkv, QKV_N, HIDDEN);
    // 2) RoPE current token
    rope_kernel<<<BATCH, 128, 0, stream>>>(qkv, context_lens, qr, kr, vcur);
    // 3) paged attention
    attn_kernel<<<BATCH * NKV, 128, 0, stream>>>(qr, kr, vcur, key_cache, value_cache,
                                                 block_tables, context_lens, attn);
    // 4) output projection: N=4096 -> 256 tiles -> 64 blocks
    gemm16_wmma<<<HIDDEN / 64, 128, 0, stream>>>(attn, W_o, out, HIDDEN, HIDDEN);
}